// SVF_34239479284429
// MI455X (gfx1250) — compile-verified
//
#include <hip/hip_runtime.h>

#define DW 160
#define DHW (DW * DW * DW)
#define NSTEP 16

#define AS1 __attribute__((address_space(1)))
#define AS3 __attribute__((address_space(3)))

typedef int v4i __attribute__((ext_vector_type(4)));
typedef AS1 v4i* g_v4i_ptr;  // global-AS int4*
typedef AS3 v4i* l_v4i_ptr;  // LDS-AS int4*

#if defined(__AMDGCN__) && __has_builtin(__builtin_amdgcn_global_load_async_to_lds_b128)
#define SVF_HAVE_ASYNC 1
#endif

// ---------------------------------------------------------------------------
// warp0 = v / 2^16, converted from planar [3,D,H,W] to AoS [D,H,W,3]
// ---------------------------------------------------------------------------
__global__ __launch_bounds__(256) void svf_init(const float* __restrict__ v,
                                                float* __restrict__ warp) {
    const int vox = blockIdx.x * 256 + threadIdx.x;
    if (vox >= DHW) return;
    const float s = 1.0f / 65536.0f;  // exact power-of-two scale
    const float a = v[vox] * s;
    const float b = v[DHW + vox] * s;
    const float c = v[2 * DHW + vox] * s;
    float* w = warp + (size_t)vox * 3;
    w[0] = a; w[1] = b; w[2] = c;
}

// ---------------------------------------------------------------------------
// One scaling-and-squaring step:
//   wout(x) = win(x) + trilinear_border(win, identity(x) + win(x))
// AoS layout, one W-row per 160-thread block (5 waves, wave32).
// ---------------------------------------------------------------------------
__global__ __launch_bounds__(DW) void svf_step(const float* __restrict__ win,
                                               float* __restrict__ wout) {
    const int x = threadIdx.x;
    const int y = blockIdx.x % DW;
    const int z = blockIdx.x / DW;
    const int vox = (z * DW + y) * DW + x;

    const float* wp = win + (size_t)vox * 3;
    const float w0 = wp[0], w1 = wp[1], w2 = wp[2];

    // CDNA5 prefetch path (global_prefetch_b8): warm the next z-slice row of
    // the perfectly-sequential own-warp stream.
    if (z + 1 < DW) __builtin_prefetch(wp + (size_t)3 * DW * DW, 0, 0);

    const float S = (float)DW;
    // identity grid, align_corners=False pixel centers (analytic, no load)
    const float ix = (2.0f * (float)x + 1.0f) / S - 1.0f;
    const float iy = (2.0f * (float)y + 1.0f) / S - 1.0f;
    const float iz = (2.0f * (float)z + 1.0f) / S - 1.0f;

    // unnormalize sample position, clamp to border
    float gx = ((ix + w0 + 1.0f) * S - 1.0f) * 0.5f;
    float gy = ((iy + w1 + 1.0f) * S - 1.0f) * 0.5f;
    float gz = ((iz + w2 + 1.0f) * S - 1.0f) * 0.5f;
    gx = fminf(fmaxf(gx, 0.0f), S - 1.0f);
    gy = fminf(fmaxf(gy, 0.0f), S - 1.0f);
    gz = fminf(fmaxf(gz, 0.0f), S - 1.0f);

    const int x0 = (int)gx, y0 = (int)gy, z0 = (int)gz;  // >=0, trunc==floor
    const float fx = gx - (float)x0;
    const float fy = gy - (float)y0;
    const float fz = gz - (float)z0;
    const int x1 = min(x0 + 1, DW - 1);
    const int y1 = min(y0 + 1, DW - 1);
    const int z1 = min(z0 + 1, DW - 1);

    const size_t r00 = (size_t)(z0 * DW + y0) * DW;
    const size_t r01 = (size_t)(z0 * DW + y1) * DW;
    const size_t r10 = (size_t)(z1 * DW + y0) * DW;
    const size_t r11 = (size_t)(z1 * DW + y1) * DW;
    const float* p[8] = {
        win + (r00 + x0) * 3, win + (r00 + x1) * 3,
        win + (r01 + x0) * 3, win + (r01 + x1) * 3,
        win + (r10 + x0) * 3, win + (r10 + x1) * 3,
        win + (r11 + x0) * 3, win + (r11 + x1) * 3,
    };
    float c[8][3];
#pragma unroll
    for (int i = 0; i < 8; ++i) {  // 8 corners x float3, AoS -> b96-mergeable
        c[i][0] = p[i][0]; c[i][1] = p[i][1]; c[i][2] = p[i][2];
    }

    float* op = wout + (size_t)vox * 3;
    float own[3] = {w0, w1, w2};
#pragma unroll
    for (int ch = 0; ch < 3; ++ch) {
        const float c00 = c[0][ch] * (1.0f - fx) + c[1][ch] * fx;
        const float c01 = c[2][ch] * (1.0f - fx) + c[3][ch] * fx;
        const float c10 = c[4][ch] * (1.0f - fx) + c[5][ch] * fx;
        const float c11 = c[6][ch] * (1.0f - fx) + c[7][ch] * fx;
        const float a  = c00 * (1.0f - fy) + c01 * fy;
        const float b  = c10 * (1.0f - fy) + c11 * fy;
        op[ch] = own[ch] + (a * (1.0f - fz) + b * fz);
    }
}

// ---------------------------------------------------------------------------
// out[c,d,h,w] = identity_c + warp[d,h,w,c]  (AoS -> planar via LDS staging,
// async-copied into LDS with the CDNA5 async-to-LDS path when available)
// ---------------------------------------------------------------------------
__global__ __launch_bounds__(DW) void svf_final(const float* __restrict__ warp,
                                                float* __restrict__ out) {
    const int x = threadIdx.x;
    const int y = blockIdx.x % DW;
    const int z = blockIdx.x / DW;
    const int vox = (z * DW + y) * DW + x;

    __shared__ __align__(16) float tile[DW * 3];  // 1920 B, 16B-aligned rows
    const float* src = warp + (size_t)blockIdx.x * (DW * 3);

#if defined(SVF_HAVE_ASYNC)
    if (x < (DW * 3) / 4) {  // 120 lanes x b128 = 1920 B
        __builtin_amdgcn_global_load_async_to_lds_b128(
            (g_v4i_ptr)(src + 4 * x), (l_v4i_ptr)(&tile[4 * x]), 0, 0);
    }
#if __has_builtin(__builtin_amdgcn_s_wait_asynccnt)
    __builtin_amdgcn_s_wait_asynccnt(0);
#else
    asm volatile("s_wait_asynccnt 0" ::: "memory");
#endif
    __syncthreads();
#else
    for (int i = x; i < DW * 3; i += DW) tile[i] = src[i];
    __syncthreads();
#endif

    const float S = (float)DW;
    out[0 * DHW + vox] = (2.0f * (float)x + 1.0f) / S - 1.0f + tile[x * 3 + 0];
    out[1 * DHW + vox] = (2.0f * (float)y + 1.0f) / S - 1.0f + tile[x * 3 + 1];
    out[2 * DHW + vox] = (2.0f * (float)z + 1.0f) / S - 1.0f + tile[x * 3 + 2];
}

// ---------------------------------------------------------------------------
extern "C" void kernel_launch(void* const* d_in, const int* in_sizes, int n_in,
                              void* d_out, int out_size, void* d_ws, size_t ws_size,
                              hipStream_t stream) {
    (void)in_sizes; (void)n_in; (void)out_size; (void)ws_size;
    // d_in[0] = identity_grid (unused: computed analytically), d_in[1] = v
    const float* v = (const float*)d_in[1];

    float* bufA = (float*)d_ws;   // AoS warp, 49.15 MB
    float* bufB = (float*)d_out;  // reuse output memory as ping-pong scratch

    svf_init<<<dim3(DHW / 256), dim3(256), 0, stream>>>(v, bufA);

    const float* cur = bufA;
    float* nxt = bufB;
    for (int s = 0; s < NSTEP; ++s) {
        svf_step<<<dim3(DW * DW), dim3(DW), 0, stream>>>(cur, nxt);
        float* t = (float*)cur; cur = nxt; nxt = t;
    }
    // NSTEP is even -> final warp lives in bufA (d_ws); safe to overwrite d_out.
    svf_final<<<dim3(DW * DW), dim3(DW), 0, stream>>>(cur, (float*)d_out);
}